// PropModel_12360915878231
// MI455X (gfx1250) — compile-verified
//
#include <hip/hip_runtime.h>

// CDNA5 / gfx1250. wave32. D = 256 throughout.
typedef __attribute__((ext_vector_type(2))) float v2f;
typedef __attribute__((ext_vector_type(8))) float v8f;

#define D 256
#define TM 64          // rows per block tile
#define LDST 260       // padded LDS row stride in floats (260*4 bytes, 16B-aligned rows)
#define WROWS 32       // K-rows of a weight matrix staged in LDS per chunk

// Load a TM x 256 row tile from global (row stride 256 floats, rows contiguous)
// into LDS (row stride LDST). Optionally duplicate into a second LDS buffer.
__device__ __forceinline__ void load_tile64(const float* __restrict__ g,
                                            float* __restrict__ dst,
                                            float* __restrict__ dst2) {
  const int t = threadIdx.x;
#pragma unroll
  for (int it = 0; it < (TM * D / 4) / 256; ++it) {  // 16 float4 per thread
    int idx = t + it * 256;       // 0..4095 float4 slots; global is contiguous
    int row = idx >> 6;           // 64 float4 per row
    int c4  = idx & 63;
    float4 v = ((const float4*)g)[idx];
    *(float4*)(dst + row * LDST + c4 * 4) = v;
    if (dst2) *(float4*)(dst2 + row * LDST + c4 * 4) = v;
  }
}

// acc += A_tile(64x256, in LDS) @ W(256x256, global, staged via lds_W chunks).
// Wave layout: band = 16-row band (0..3), half = 128-col half (0..1).
// Per wave: 8 output tiles of 16x16 (v8f accumulators).
// WMMA f32 16x16x4 fragment layout (ISA 7.12.2):
//   A: lane m=l%16 -> row M=m, VGPR v -> K = 2*(l/16)+v
//   B: VGPR v, lane l -> K = 2*(l/16)+v, N = l%16
//   C/D: VGPR r, lane l -> M = r + 8*(l/16), N = l%16
__device__ __forceinline__ void tile_gemm(const float* __restrict__ lds_A,
                                          const float* __restrict__ Wg,
                                          float* __restrict__ lds_W,
                                          v8f acc[8],
                                          int band, int half, int m, int hi) {
  const float* arow = lds_A + (band * 16 + m) * LDST;
  for (int chunk = 0; chunk < D / WROWS; ++chunk) {
    __syncthreads();  // previous users of lds_W are done
    {
      const float4* src = (const float4*)(Wg + chunk * WROWS * D);
      const int t = threadIdx.x;
#pragma unroll
      for (int it = 0; it < (WROWS * D / 4) / 256; ++it) {  // 8 float4 per thread
        int idx = t + it * 256;
        int row = idx >> 6;
        int c4  = idx & 63;
        *(float4*)(lds_W + row * LDST + c4 * 4) = src[idx];
      }
    }
    __syncthreads();
#pragma unroll
    for (int k0 = 0; k0 < WROWS; k0 += 4) {
      v2f a;
      a.x = arow[chunk * WROWS + k0 + 2 * hi];
      a.y = arow[chunk * WROWS + k0 + 2 * hi + 1];
      const float* w0 = lds_W + (k0 + 2 * hi) * LDST + half * 128 + m;
#pragma unroll
      for (int ct = 0; ct < 8; ++ct) {
        v2f b;
        b.x = w0[ct * 16];
        b.y = w0[ct * 16 + LDST];
        acc[ct] = __builtin_amdgcn_wmma_f32_16x16x4_f32(
            false, a, false, b, (short)0, acc[ct], false, false);
      }
    }
  }
  __syncthreads();  // all A/W reads complete before caller reuses buffers
}

__device__ __forceinline__ void zero8(v8f acc[8]) {
#pragma unroll
  for (int c = 0; c < 8; ++c) acc[c] = (v8f){};
}

__global__ __launch_bounds__(256) void prop_chain_kernel(
    const float* __restrict__ user_emb,  // [4][N][256]
    const float* __restrict__ Wv, const float* __restrict__ bv,
    const float* __restrict__ Wo, const float* __restrict__ bo,
    const float* __restrict__ prop_w,    // [6][256][256]
    float* __restrict__ out,             // [4][N][256]
    int N) {
  __shared__ __align__(16) float lds_E[TM * LDST];   // E_i tile (immutable)
  __shared__ __align__(16) float lds_C[TM * LDST];   // carry, then M = carry*E_j
  __shared__ __align__(16) float lds_B[TM * LDST];   // E_j tile
  __shared__ __align__(16) float lds_H[TM * LDST];   // H = E_j@Wv+bv
  __shared__ __align__(16) float lds_W[WROWS * LDST];

  const int i = blockIdx.y + 1;                 // output type 1..3
  const long long rowbase = (long long)blockIdx.x * TM;
  const long long ND = (long long)N * D;

  const int lane = threadIdx.x & 31;
  const int wave = threadIdx.x >> 5;
  const int band = wave >> 1;                   // 0..3  (16-row band)
  const int half = wave & 1;                    // 0..1  (128-col half)
  const int m    = lane & 15;
  const int hi   = lane >> 4;
  const int rowoff  = band * 16 + 8 * hi;       // + r (0..7) = row in tile
  const int colbase = half * 128 + m;           // + ct*16    = col in tile

  // carry = E_i (duplicated so the base term survives the chain)
  load_tile64(user_emb + (long long)i * ND + rowbase * D, lds_E, lds_C);

  for (int j = 0; j < i; ++j) {
    __syncthreads();  // prev-iteration readers of lds_B / writers of lds_C done
    load_tile64(user_emb + (long long)j * ND + rowbase * D, lds_B, nullptr);

    // ---- score S_j = (E_j @ Wv + bv) @ Wo + bo  (attention collapses to V) ----
    v8f acc[8];
    zero8(acc);
    tile_gemm(lds_B, Wv, lds_W, acc, band, half, m, hi);
#pragma unroll
    for (int ct = 0; ct < 8; ++ct) {
      float bvv = bv[colbase + ct * 16];
#pragma unroll
      for (int r = 0; r < 8; ++r)
        lds_H[(rowoff + r) * LDST + colbase + ct * 16] = acc[ct][r] + bvv;
    }
    __syncthreads();

    v8f sacc[8];
    zero8(sacc);
    tile_gemm(lds_H, Wo, lds_W, sacc, band, half, m, hi);
#pragma unroll
    for (int ct = 0; ct < 8; ++ct) {
      float bov = bo[colbase + ct * 16];
#pragma unroll
      for (int r = 0; r < 8; ++r) sacc[ct][r] += bov;
    }

    // ---- M = carry ⊙ E_j (in place; per-element ownership, no race) ----
#pragma unroll
    for (int it = 0; it < TM * D / 256; ++it) {  // 64 elements per thread
      int idx = threadIdx.x + it * 256;
      int row = idx >> 8;
      int c   = idx & 255;
      lds_C[row * LDST + c] *= lds_B[row * LDST + c];
    }
    // tile_gemm's leading barrier makes M visible before A-fragment reads.

    // ---- T = M @ prop_w[i-1+j] ----
    v8f tacc[8];
    zero8(tacc);
    tile_gemm(lds_C, prop_w + (long long)(i - 1 + j) * D * D, lds_W, tacc,
              band, half, m, hi);
    // trailing barrier in tile_gemm: all reads of lds_C done -> safe to overwrite

    // ---- final = E_i + leaky(T * S);  carry <- final; last j writes out ----
    float* outp = out + (long long)i * ND + rowbase * D;
    const bool last = (j == i - 1);
#pragma unroll
    for (int ct = 0; ct < 8; ++ct) {
#pragma unroll
      for (int r = 0; r < 8; ++r) {
        int rr = rowoff + r;
        int cc = colbase + ct * 16;
        float t = tacc[ct][r] * sacc[ct][r];
        float mes = (t >= 0.0f) ? t : 0.01f * t;
        float f = lds_E[rr * LDST + cc] + mes;
        lds_C[rr * LDST + cc] = f;
        if (last) outp[(long long)rr * D + cc] = f;
      }
    }
  }
}

extern "C" void kernel_launch(void* const* d_in, const int* in_sizes, int n_in,
                              void* d_out, int out_size, void* d_ws, size_t ws_size,
                              hipStream_t stream) {
  (void)n_in; (void)out_size; (void)d_ws; (void)ws_size;
  // setup_inputs order: user_emb, Wq, bq, Wk, bk, Wv, bv, Wo, bo, prop_w
  const float* user_emb = (const float*)d_in[0];
  const float* Wv = (const float*)d_in[5];
  const float* bv = (const float*)d_in[6];
  const float* Wo = (const float*)d_in[7];
  const float* bo = (const float*)d_in[8];
  const float* prop_w = (const float*)d_in[9];
  float* out = (float*)d_out;

  const int N = in_sizes[0] / (4 * D);  // 65536

  // out[0] = user_emb[0]
  hipMemcpyAsync(out, user_emb, (size_t)N * D * sizeof(float),
                 hipMemcpyDeviceToDevice, stream);

  dim3 grid(N / TM, 3, 1);
  prop_chain_kernel<<<grid, 256, 0, stream>>>(user_emb, Wv, bv, Wo, bo, prop_w,
                                              out, N);
}